// LowLevelModel_87771951661557
// MI455X (gfx1250) — compile-verified
//
#include <hip/hip_runtime.h>
#include <hip/hip_bf16.h>

#define B_    16
#define NAV_  36
#define HIM_  120
#define WIM_  160
#define RH_   512
#define NF_   4
#define AE_   37
#define KC1_  589            // RH+NAV+NF+AE
#define KCAT_ 1101           // KC1 + RH
#define KPAD_ 1104           // padded to multiple of 4
#define KSTEPS_ (KPAD_/4)    // 276 WMMA K-steps
#define NTILE_ 128           // GDIM/16
#define GDIM_ 2048           // 4*RH
#define PANO_H_ 360
#define PANO_W_ 1920
#define HW_   (HIM_*WIM_)

typedef float v2f __attribute__((ext_vector_type(2)));
typedef float v8f __attribute__((ext_vector_type(8)));

// ---------------- workspace layout (floats) ----------------
static const size_t PD_OFF    = 0;                                   // 16*36*120*160
static const size_t PANO_OFF  = PD_OFF   + (size_t)B_*NAV_*HW_;      // 16*360*1920
static const size_t ATTN_OFF  = PANO_OFF + (size_t)B_*PANO_H_*PANO_W_;
static const size_t DFA_OFF   = ATTN_OFF + B_*NAV_;
static const size_t KERN_OFF  = DFA_OFF  + B_*NF_;
static const size_t APACK_OFF = KERN_OFF + (size_t)B_*NAV_*50;       // 276*64
static const size_t BPACK_OFF = APACK_OFF + (size_t)KSTEPS_*64;      // 128*276*64
static const size_t GATES_OFF = BPACK_OFF + (size_t)NTILE_*KSTEPS_*64;
static const size_t VF_OFF    = GATES_OFF + (size_t)B_*GDIM_;

// ---------------- kernel 1: attn softmax, dyn MLP, filter mix ----------------
__global__ void k_prep(const float* __restrict__ nnf, const float* __restrict__ ctx,
                       const float* __restrict__ fc1w, const float* __restrict__ fc1b,
                       const float* __restrict__ fc2w, const float* __restrict__ fc2b,
                       const float* __restrict__ dynf,
                       float* __restrict__ attn, float* __restrict__ dfa,
                       float* __restrict__ kern) {
  __shared__ float hid[B_][RH_/4];
  __shared__ float dfas[B_][NF_];
  const int tid = threadIdx.x;
  if (tid < B_) {                                        // attn = softmax over 36
    float m = -1e30f;
    for (int c = 0; c < NAV_; ++c) m = fmaxf(m, nnf[tid*NAV_ + c]);
    float e[NAV_]; float s = 0.f;
    for (int c = 0; c < NAV_; ++c) { e[c] = __expf(nnf[tid*NAV_ + c] - m); s += e[c]; }
    float inv = 1.f / s;
    for (int c = 0; c < NAV_; ++c) attn[tid*NAV_ + c] = e[c] * inv;
  }
  for (int it = tid; it < B_*(RH_/4); it += blockDim.x) { // hid = relu(ctx @ fc1^T + b1)
    int b = it / (RH_/4), j = it % (RH_/4);
    float acc = fc1b[j];
    const float* cb = ctx + b*RH_;
    const float* wr = fc1w + j*RH_;
    for (int k = 0; k < RH_; ++k) acc = __builtin_fmaf(cb[k], wr[k], acc);
    hid[b][j] = fmaxf(acc, 0.f);
  }
  __syncthreads();
  if (tid < B_) {                                        // dfa = softmax(hid @ fc2^T + b2)
    float lg[NF_]; float m = -1e30f;
    for (int f = 0; f < NF_; ++f) {
      float a = fc2b[f];
      for (int j = 0; j < RH_/4; ++j) a = __builtin_fmaf(hid[tid][j], fc2w[f*(RH_/4)+j], a);
      lg[f] = a; m = fmaxf(m, a);
    }
    float s = 0.f;
    for (int f = 0; f < NF_; ++f) { lg[f] = __expf(lg[f] - m); s += lg[f]; }
    float inv = 1.f / s;
    for (int f = 0; f < NF_; ++f) { dfas[tid][f] = lg[f]*inv; dfa[tid*NF_ + f] = lg[f]*inv; }
  }
  __syncthreads();
  for (int it = tid; it < B_*NAV_*50; it += blockDim.x) { // kern = dfa @ filter bank
    int b = it / (NAV_*50), r = it % (NAV_*50);
    float a = 0.f;
    for (int f = 0; f < NF_; ++f) a = __builtin_fmaf(dfas[b][f], dynf[f*NAV_*50 + r], a);
    kern[it] = a;
  }
}

// ---------------- kernel 2: grouped conv1 fused with attn scale -> pd ----------------
__global__ void k_conv1(const float* __restrict__ draw, const float* __restrict__ dclip,
                        const float* __restrict__ w, const float* __restrict__ bias,
                        const float* __restrict__ attn, float* __restrict__ pd) {
  long idx = (long)blockIdx.x*blockDim.x + threadIdx.x;
  if (idx >= (long)B_*NAV_*HW_) return;
  int x = (int)(idx % WIM_); long t = idx / WIM_;
  int y = (int)(t % HIM_); t /= HIM_;
  int g = (int)(t % NAV_); int b = (int)(t / NAV_);
  const float* wr = w + g*50;
  const float* p0 = draw  + ((long)b*NAV_ + g)*HW_;
  const float* p1 = dclip + ((long)b*NAV_ + g)*HW_;
  float acc = bias[g];
#pragma unroll
  for (int ky = 0; ky < 5; ++ky) {
    int yy = y + ky - 2;
    if (yy < 0 || yy >= HIM_) continue;
#pragma unroll
    for (int kx = 0; kx < 5; ++kx) {
      int xx = x + kx - 2;
      if (xx < 0 || xx >= WIM_) continue;
      long o = (long)yy*WIM_ + xx;
      acc = __builtin_fmaf(p0[o], wr[ky*5 + kx], acc);
      acc = __builtin_fmaf(p1[o], wr[25 + ky*5 + kx], acc);
    }
  }
  pd[idx] = acc * attn[b*NAV_ + g];
}

// ---------------- kernel 3: dynamic grouped conv -> pano layout ----------------
__global__ void k_dynconv(const float* __restrict__ pd, const float* __restrict__ obj,
                          const float* __restrict__ attn, const float* __restrict__ kern,
                          float* __restrict__ pano) {
  long idx = (long)blockIdx.x*blockDim.x + threadIdx.x;
  if (idx >= (long)B_*NAV_*HW_) return;
  int x = (int)(idx % WIM_); long t = idx / WIM_;
  int y = (int)(t % HIM_); t /= HIM_;
  int g = (int)(t % NAV_); int b = (int)(t / NAV_);
  const float* kb = kern + ((long)b*NAV_ + g)*50;
  const float* p0 = pd  + ((long)b*NAV_ + g)*HW_;
  const float* p1 = obj + ((long)b*NAV_ + g)*HW_;
  const float av = attn[b*NAV_ + g];
  float acc = 0.f;
#pragma unroll
  for (int ky = 0; ky < 5; ++ky) {
    int yy = y + ky - 2;
    if (yy < 0 || yy >= HIM_) continue;
#pragma unroll
    for (int kx = 0; kx < 5; ++kx) {
      int xx = x + kx - 2;
      if (xx < 0 || xx >= WIM_) continue;
      long o = (long)yy*WIM_ + xx;
      acc = __builtin_fmaf(p0[o],      kb[ky*5 + kx],      acc);
      acc = __builtin_fmaf(p1[o] * av, kb[25 + ky*5 + kx], acc);
    }
  }
  int r = 2 - g/12, c = g % 12;                 // pano[:, ::-1] row flip
  pano[((long)b*PANO_H_ + (long)r*HIM_ + y)*PANO_W_ + (long)c*WIM_ + x] = acc;
}

// ---------------- kernel 4: panorama conv chain (both roll variants) -> vf ----------------
__global__ void k_conv2path(const float* __restrict__ pano,
                            const float* __restrict__ w21, const float* __restrict__ b21p,
                            const float* __restrict__ w22, const float* __restrict__ b22p,
                            float* __restrict__ vf_ws, float* __restrict__ vf_out) {
  __shared__ float s2[2][21][89];
  __shared__ float s3[2][10][43];
  __shared__ float s4[2][36];
  __shared__ float wk1[25], wk2[9];
  const int b = blockIdx.x, tid = threadIdx.x;
  if (tid < 25) wk1[tid] = w21[tid];
  if (tid < 9)  wk2[tid] = w22[tid];
  __syncthreads();
  const float bias1 = b21p[0], bias2 = b22p[0];
  const float* P = pano + (long)b*PANO_H_*PANO_W_;
  // conv2_1 (5x5, stride 5x7, dil 10x10) + avgpool 3x3/3x3
  for (int it = tid; it < 2*21*89; it += blockDim.x) {
    int v = it / (21*89), rem = it % (21*89);
    int p = rem / 89, q = rem % 89;
    float acc = 0.f;
    for (int a = 0; a < 3; ++a)
      for (int bb = 0; bb < 3; ++bb) {
        int i = 3*p + a, j = 3*q + bb;
        float c1 = bias1;
#pragma unroll
        for (int u = 0; u < 5; ++u)
#pragma unroll
          for (int vv = 0; vv < 5; ++vv) {
            int row = i*5 + u*10;
            int col = j*7 + vv*10;
            if (v) col = (col + (PANO_W_ - WIM_)) % PANO_W_;   // roll(pano, +WIM)
            c1 = __builtin_fmaf(P[(long)row*PANO_W_ + col], wk1[u*5 + vv], c1);
          }
        acc += c1;
      }
    s2[v][p][q] = acc * (1.f/9.f);
  }
  __syncthreads();
  // conv2_2 (3x3, stride 2x2, dil 1x2)
  for (int it = tid; it < 2*10*43; it += blockDim.x) {
    int v = it / (10*43), rem = it % (10*43);
    int m = rem / 43, n = rem % 43;
    float acc = bias2;
#pragma unroll
    for (int u = 0; u < 3; ++u)
#pragma unroll
      for (int w3 = 0; w3 < 3; ++w3)
        acc = __builtin_fmaf(s2[v][2*m + u][2*n + 2*w3], wk2[u*3 + w3], acc);
    s3[v][m][n] = acc;
  }
  __syncthreads();
  // avgpool 3x9 / 3x3 -> [2][3][12]
  for (int it = tid; it < 2*36; it += blockDim.x) {
    int v = it / 36, rem = it % 36;
    int r = rem / 12, s = rem % 12;
    float acc = 0.f;
    for (int a = 0; a < 3; ++a)
      for (int bb = 0; bb < 9; ++bb) acc += s3[v][3*r + a][3*s + bb];
    s4[v][rem] = acc * (1.f/27.f);
  }
  __syncthreads();
  if (tid == 0) {   // combine variants, row flip, softmax(vf/10)
    float vfl[36]; float m = -1e30f;
    for (int k = 0; k < 36; ++k) {
      int r = 2 - k/12, s = k % 12;
      float val = 0.5f * (s4[0][r*12 + s] + s4[1][r*12 + (s + 1) % 12]);
      vfl[k] = val * 0.1f;
      m = fmaxf(m, vfl[k]);
    }
    float ssum = 0.f;
    for (int k = 0; k < 36; ++k) { vfl[k] = __expf(vfl[k] - m); ssum += vfl[k]; }
    float inv = 1.f / ssum;
    for (int k = 0; k < 36; ++k) {
      float o = vfl[k] * inv;
      vf_ws[b*36 + k]  = o;
      vf_out[b*36 + k] = o;
    }
  }
}

// ---------------- kernel 5: build A in WMMA-fragment layout; zero nav_mask ----------------
// Apack[s*64 + lane*2 + j] = A[m = lane%16][k = 4s + (lane/16)*2 + j]
__global__ void k_concat(const float* __restrict__ ctx, const float* __restrict__ vf,
                         const float* __restrict__ dfa, const float* __restrict__ pre,
                         const float* __restrict__ h0,
                         float* __restrict__ Apack, float* __restrict__ mask) {
  int idx = blockIdx.x*blockDim.x + threadIdx.x;
  if (idx < KSTEPS_*64) {
    int s = idx >> 6, r = idx & 63;
    int lane = r >> 1, j = r & 1;
    int b = lane & 15;
    int k = 4*s + (lane >> 4)*2 + j;
    float v = 0.f;
    if      (k < RH_)          v = ctx[b*RH_ + k];
    else if (k < RH_+NAV_)     v = vf[b*NAV_ + (k - RH_)];
    else if (k < RH_+NAV_+NF_) v = dfa[b*NF_ + (k - RH_ - NAV_)];
    else if (k < KC1_)         v = pre[b*AE_ + (k - RH_ - NAV_ - NF_)];
    else if (k < KCAT_)        v = h0[b*RH_ + (k - KC1_)];
    Apack[idx] = v;
  } else if (idx < KSTEPS_*64 + B_*NAV_) {
    mask[idx - KSTEPS_*64] = 0.f;
  }
}

__global__ void k_scatter(const int* __restrict__ nidx, float* __restrict__ mask) {
  int t = threadIdx.x;
  if (t < B_*8) {
    int b = t / 8;
    mask[b*NAV_ + nidx[t]] = 1.0f;
  }
}

// ---------------- kernel 5b: pack W = [wih | whh | 0] into WMMA-fragment layout ----------------
// Bpack[((nt*KSTEPS + s)*32 + lane)*2 + j] = W[n = nt*16 + lane%16][k = 4s + (lane/16)*2 + j]
__global__ void k_packB(const float* __restrict__ wih, const float* __restrict__ whh,
                        float* __restrict__ Bpack) {
  long idx = (long)blockIdx.x*blockDim.x + threadIdx.x;
  if (idx >= (long)NTILE_*KSTEPS_*64) return;
  int j    = (int)(idx & 1);
  int lane = (int)((idx >> 1) & 31);
  long t   = idx >> 6;
  int s    = (int)(t % KSTEPS_);
  int nt   = (int)(t / KSTEPS_);
  int n = nt*16 + (lane & 15);
  int k = 4*s + (lane >> 4)*2 + j;
  float v = 0.f;
  if      (k < KC1_)  v = wih[(long)n*KC1_ + k];
  else if (k < KCAT_) v = whh[(long)n*RH_ + (k - KC1_)];
  Bpack[idx] = v;
}

// ---------------- kernel 6: LSTM gate GEMM via fp32 WMMA (M=16, N=2048, K=1104) ----------------
// Branchless inner loop: one coalesced b64 load from Apack, one from Bpack, one WMMA.
__global__ void __launch_bounds__(128) k_gates(const float* __restrict__ Apack,
                                               const float* __restrict__ Bpack,
                                               const float* __restrict__ bih,
                                               const float* __restrict__ bhh,
                                               float* __restrict__ gates) {
  const int wave = blockIdx.x * (blockDim.x >> 5) + (threadIdx.x >> 5);
  const int lane = threadIdx.x & 31;
  const int ln   = lane & 15;
  const int hi   = lane >> 4;
  const int n    = wave*16 + ln;
  v8f c = {};
  const v2f* __restrict__ Ap = (const v2f*)Apack + lane;                       // + s*32
  const v2f* __restrict__ Bp = (const v2f*)Bpack + (long)wave*KSTEPS_*32 + lane;
#pragma unroll 4
  for (int s = 0; s < KSTEPS_; ++s) {
    v2f a = Ap[(long)s*32];
    v2f b = Bp[(long)s*32];
    c = __builtin_amdgcn_wmma_f32_16x16x4_f32(false, a, false, b, (short)0, c,
                                              false, false);
  }
  const float bsum = bih[n] + bhh[n];
#pragma unroll
  for (int v = 0; v < 8; ++v) {
    int m = v + hi*8;                  // C/D layout: VGPR v -> M=v (lanes 0-15), v+8 (16-31)
    gates[(long)m*GDIM_ + n] = c[v] + bsum;
  }
}

// ---------------- kernel 7: LSTM nonlinearity -> h1, c1 ----------------
__global__ void k_lstm(const float* __restrict__ gates, const float* __restrict__ c0,
                       float* __restrict__ out) {
  int idx = blockIdx.x*blockDim.x + threadIdx.x;
  if (idx >= B_*RH_) return;
  int b = idx / RH_, j = idx % RH_;
  const float* gr = gates + (long)b*GDIM_;
  float ig = gr[j], fg = gr[RH_ + j], gg = gr[2*RH_ + j], og = gr[3*RH_ + j];
  float si = 1.f/(1.f + __expf(-ig));
  float sf = 1.f/(1.f + __expf(-fg));
  float so = 1.f/(1.f + __expf(-og));
  float c1 = sf * c0[idx] + si * tanhf(gg);
  out[idx]           = so * tanhf(c1);   // h_1
  out[B_*RH_ + idx]  = c1;               // c_1
}

// ---------------- launch ----------------
extern "C" void kernel_launch(void* const* d_in, const int* in_sizes, int n_in,
                              void* d_out, int out_size, void* d_ws, size_t ws_size,
                              hipStream_t stream) {
  const float* depth_raw = (const float*)d_in[0];
  const float* depth_clip= (const float*)d_in[1];
  const float* obj_feat  = (const float*)d_in[2];
  const float* nnf       = (const float*)d_in[3];
  const float* pre_act   = (const float*)d_in[4];
  const float* h_0       = (const float*)d_in[5];
  const float* c_0       = (const float*)d_in[6];
  const float* ctx       = (const float*)d_in[7];
  const int*   nav_idx   = (const int*)  d_in[8];
  const float* conv1_w   = (const float*)d_in[9];
  const float* conv1_b   = (const float*)d_in[10];
  const float* fc1w      = (const float*)d_in[11];
  const float* fc1b      = (const float*)d_in[12];
  const float* fc2w      = (const float*)d_in[13];
  const float* fc2b      = (const float*)d_in[14];
  const float* dynf      = (const float*)d_in[15];
  const float* w21       = (const float*)d_in[16];
  const float* b21       = (const float*)d_in[17];
  const float* w22       = (const float*)d_in[18];
  const float* b22       = (const float*)d_in[19];
  const float* wih       = (const float*)d_in[20];
  const float* whh       = (const float*)d_in[21];
  const float* bih       = (const float*)d_in[22];
  const float* bhh       = (const float*)d_in[23];

  float* ws    = (float*)d_ws;
  float* pd    = ws + PD_OFF;
  float* pano  = ws + PANO_OFF;
  float* attn  = ws + ATTN_OFF;
  float* dfa   = ws + DFA_OFF;
  float* kern  = ws + KERN_OFF;
  float* Apack = ws + APACK_OFF;
  float* Bpack = ws + BPACK_OFF;
  float* gates = ws + GATES_OFF;
  float* vf_ws = ws + VF_OFF;

  float* out   = (float*)d_out;
  float* vf_out   = out + 2*B_*RH_;           // 16384
  float* mask_out = out + 2*B_*RH_ + B_*NAV_; // 16960

  k_prep<<<1, 256, 0, stream>>>(nnf, ctx, fc1w, fc1b, fc2w, fc2b, dynf, attn, dfa, kern);

  // B-matrix pack can run early (depends only on weights)
  const long bpack_total = (long)NTILE_*KSTEPS_*64;
  k_packB<<<(int)((bpack_total + 255)/256), 256, 0, stream>>>(wih, whh, Bpack);

  const long conv_total = (long)B_*NAV_*HW_;
  const int conv_blocks = (int)((conv_total + 255) / 256);
  k_conv1<<<conv_blocks, 256, 0, stream>>>(depth_raw, depth_clip, conv1_w, conv1_b, attn, pd);
  k_dynconv<<<conv_blocks, 256, 0, stream>>>(pd, obj_feat, attn, kern, pano);

  k_conv2path<<<B_, 256, 0, stream>>>(pano, w21, b21, w22, b22, vf_ws, vf_out);

  const int cat_total = KSTEPS_*64 + B_*NAV_;
  k_concat<<<(cat_total + 255)/256, 256, 0, stream>>>(ctx, vf_ws, dfa, pre_act, h_0,
                                                      Apack, mask_out);
  k_scatter<<<1, 128, 0, stream>>>(nav_idx, mask_out);

  k_gates<<<NTILE_/4, 128, 0, stream>>>(Apack, Bpack, bih, bhh, gates);

  k_lstm<<<(B_*RH_ + 255)/256, 256, 0, stream>>>(gates, c_0, out);
}